// CLAMP_8151847928390
// MI455X (gfx1250) — compile-verified
//
#include <hip/hip_runtime.h>
#include <hip/hip_bf16.h>
#include <stdint.h>

#define NALGOS 64
#define NTASKS 1024
#define NSTEPS 512

typedef float v2f __attribute__((ext_vector_type(2)));
typedef float v8f __attribute__((ext_vector_type(8)));
typedef unsigned int u32x4 __attribute__((ext_vector_type(4)));
typedef int i32x4 __attribute__((ext_vector_type(4)));
typedef int i32x8 __attribute__((ext_vector_type(8)));

#if defined(__HIP_DEVICE_COMPILE__) && __has_builtin(__builtin_amdgcn_tensor_load_to_lds) && __has_builtin(__builtin_amdgcn_s_wait_tensorcnt)
#define HAVE_TDM 1
#else
#define HAVE_TDM 0
#endif

// sig2m1(y) computes 2*sigmoid(x)-1 = tanh(x/2) for x = result/diff, where the
// caller passes y = x * SIG_SCALE (scale folded into a precomputed per-lane constant).
#if defined(__HIP_DEVICE_COMPILE__) && __has_builtin(__builtin_amdgcn_tanhf)
#define SIG_SCALE 0.5f                       // y = x/2 -> tanh(y), single trans op
__device__ __forceinline__ float sig2m1(float y) {
    return __builtin_amdgcn_tanhf(y);
}
#else
#define SIG_SCALE 1.4426950408889634f        // y = x*log2(e) -> 1 - 2/(2^y + 1)
__device__ __forceinline__ float sig2m1(float y) {
    float e = __builtin_amdgcn_exp2f(y);
    return 1.0f - 2.0f * __builtin_amdgcn_rcpf(e + 1.0f);
}
#endif

#if HAVE_TDM
// DMA one 4KB row (1024 f32) from global memory into LDS via the Tensor Data Mover.
__device__ __forceinline__ void tdm_row_load(const float* gsrc, void* lds_dst) {
    uint64_t ga = (uint64_t)(uintptr_t)gsrc;
    uint32_t la = (uint32_t)(uintptr_t)lds_dst;  // generic LDS addr low 32 bits = LDS byte offset
    u32x4 g0;
    g0[0] = 1u;                                                 // count=1, user descriptor
    g0[1] = la;                                                 // lds_addr
    g0[2] = (uint32_t)ga;                                       // global_addr[31:0]
    g0[3] = ((uint32_t)(ga >> 32) & 0x01FFFFFFu) | 0x80000000u; // global_addr[56:32] | type=2
    i32x8 g1;
    g1[0] = (int)(2u << 16);        // workgroup_mask=0, data_size=2 (4 bytes)
    g1[1] = (int)(1024u << 16);     // tensor_dim0[15:0] = 1024
    g1[2] = (int)(1u << 16);        // tensor_dim0[31:16]=0, tensor_dim1[15:0]=1
    g1[3] = (int)(1024u << 16);     // tensor_dim1[31:16]=0, tile_dim0=1024
    g1[4] = 1;                      // tile_dim1=1, tile_dim2=0
    g1[5] = 1024;                   // tensor_dim0_stride[31:0] = 1024
    g1[6] = (int)(1024u << 16);     // tensor_dim1_stride[15:0]
    g1[7] = 0;
    i32x4 g2 = {0, 0, 0, 0};
    i32x4 g3 = {0, 0, 0, 0};
#if __clang_major__ >= 23
    i32x8 g4 = {0, 0, 0, 0, 0, 0, 0, 0};
    __builtin_amdgcn_tensor_load_to_lds(g0, g1, g2, g3, g4, 0);
#else
    __builtin_amdgcn_tensor_load_to_lds(g0, g1, g2, g3, 0);
#endif
}
#endif

__global__ __launch_bounds__(1024, 1)
void scan_perf_kernel(const int* __restrict__ lx,
                      const float* __restrict__ tm,
                      const float* __restrict__ diff,
                      const float* __restrict__ eff,
                      const float* __restrict__ memv,
                      const float* __restrict__ boost,
                      float* __restrict__ outp) {
    __shared__ float rowLDS[2][NTASKS];   // double-buffered current task row
    __shared__ float perfS[2][NALGOS];    // double-buffered raw result[:, lx[t]] column

    const int tid      = threadIdx.x;        // 0..1023
    const int wave     = tid >> 5;           // 0..31
    const int lane     = tid & 31;
    const int lanelow  = lane & 15;
    const int hi8      = (lane >> 4) * 8;    // 0 or 8
    const int ac       = wave & 3;           // algo chunk (16 algos)
    const int ac16     = ac * 16;
    const int colgroup = wave >> 2;          // 0..7, owns columns [colgroup*128, +128)
    const int colbase  = colgroup * 128;

    // Per-lane constants
    const float eff_l   = eff[ac16 + lanelow];
    const float boost_l = boost[ac16 + lanelow];
    float mem_r[8];
    float scl_col[8];        // SIG_SCALE / diff[n_j] for this lane's column in tile j
    unsigned base_j[8];      // output base index for (a0, n_j)
#pragma unroll
    for (int r = 0; r < 8; ++r) mem_r[r] = memv[ac16 + hi8 + r];
#pragma unroll
    for (int j = 0; j < 8; ++j) {
        const int nj = colbase + j * 16 + lanelow;
        scl_col[j] = SIG_SCALE * __builtin_amdgcn_rcpf(diff[nj]);
        base_j[j]  = (unsigned)((ac16 + hi8) * NTASKS + nj) * 513u;
    }

    // Zero plane at t=0 of the output
#pragma unroll
    for (int j = 0; j < 8; ++j)
#pragma unroll
        for (int r = 0; r < 8; ++r)
            outp[(size_t)base_j[j] + (size_t)r * 525312u] = 0.0f;

    // result state: 8 tiles of 16x16 f32 held in WMMA accumulator layout
    v8f D[8];
#pragma unroll
    for (int j = 0; j < 8; ++j) D[j] = (v8f)0.0f;

    if (tid < NALGOS) perfS[0][tid] = 0.0f;   // sig_{-1} column is zero

    // Prologue: fetch row for step 0
    int task = __builtin_amdgcn_readfirstlane(lx[0]);
#if HAVE_TDM
    if (tid < 32) {
        tdm_row_load(tm + (size_t)task * NTASKS, (void*)&rowLDS[0][0]);
        __builtin_amdgcn_s_wait_tensorcnt(0);
    }
#else
    rowLDS[0][tid] = tm[(size_t)task * NTASKS + tid];
#endif
    __syncthreads();

    for (int t = 0; t < NSTEPS; ++t) {
        const int buf = t & 1;
        const int nb  = buf ^ 1;
        const bool havenext = (t + 1) < NSTEPS;
        const int  task1    = havenext ? __builtin_amdgcn_readfirstlane(lx[t + 1]) : 0;

        // Kick async TDM fetch of next step's row (overlaps with everything below)
#if HAVE_TDM
        if (havenext && tid < 32)
            tdm_row_load(tm + (size_t)task1 * NTASKS, (void*)&rowLDS[nb][0]);
#else
        if (havenext)
            rowLDS[nb][tid] = tm[(size_t)task1 * NTASKS + tid];
#endif

        // prior_perf[a] = 2*sigmoid(result_{t-1}[a,task]/diff[task])-1; raw value staged last step
        const float scl_task = SIG_SCALE * __builtin_amdgcn_rcpf(diff[task]);
        const float praw  = perfS[buf][ac16 + lanelow];
        const float pv    = sig2m1(praw * scl_task);
        const float coeff = 0.25f * (eff_l + pv * boost_l);  // 0.25: A replicated over K=4
        v2f A; A[0] = coeff; A[1] = coeff;

        // Load all 8 row segments first so the DS loads pipeline ahead of the WMMAs
        float brow[8];
#pragma unroll
        for (int j = 0; j < 8; ++j)
            brow[j] = rowLDS[buf][colbase + j * 16 + lanelow];

        // result = result*mem + coeff (x) row   via  D = A x B + C
#pragma unroll
        for (int j = 0; j < 8; ++j) {
            v2f B; B[0] = brow[j]; B[1] = brow[j];
            v8f C;
#pragma unroll
            for (int r = 0; r < 8; ++r) C[r] = D[j][r] * mem_r[r];
            D[j] = __builtin_amdgcn_wmma_f32_16x16x4_f32(
                false, A, false, B, (short)0, C, false, false);
        }

        // Stage raw result[:, lx[t+1]] column (2 active lanes per owning wave)
        if (havenext) {
            const int ctile = task1 >> 4;
            if ((ctile >> 3) == colgroup && lanelow == (task1 & 15)) {
                v8f dv;
                switch (ctile & 7) {
                    case 0: dv = D[0]; break; case 1: dv = D[1]; break;
                    case 2: dv = D[2]; break; case 3: dv = D[3]; break;
                    case 4: dv = D[4]; break; case 5: dv = D[5]; break;
                    case 6: dv = D[6]; break; default: dv = D[7]; break;
                }
#pragma unroll
                for (int r = 0; r < 8; ++r)
                    perfS[nb][ac16 + hi8 + r] = dv[r];
            }
        }

        // sig = 2*sigmoid(result/diff[n])-1 -> out[a][n][t+1]
#pragma unroll
        for (int j = 0; j < 8; ++j) {
#pragma unroll
            for (int r = 0; r < 8; ++r) {
                const float s = sig2m1(D[j][r] * scl_col[j]);
                outp[(size_t)base_j[j] + (size_t)r * 525312u + (size_t)(t + 1)] = s;
            }
        }

        task = task1;
#if HAVE_TDM
        if (tid < 32) __builtin_amdgcn_s_wait_tensorcnt(0);
#endif
        __syncthreads();
    }
}

extern "C" void kernel_launch(void* const* d_in, const int* in_sizes, int n_in,
                              void* d_out, int out_size, void* d_ws, size_t ws_size,
                              hipStream_t stream) {
    const int*   lx    = (const int*)d_in[0];
    const float* tm    = (const float*)d_in[1];
    const float* diff  = (const float*)d_in[2];
    const float* eff   = (const float*)d_in[3];
    const float* memv  = (const float*)d_in[4];
    const float* boost = (const float*)d_in[5];
    float* out = (float*)d_out;
    hipLaunchKernelGGL(scan_perf_kernel, dim3(1), dim3(1024), 0, stream,
                       lx, tm, diff, eff, memv, boost, out);
}